// SparseMaskHead_12446815224510
// MI455X (gfx1250) — compile-verified
//
#include <hip/hip_runtime.h>

// ---------------- problem constants ----------------
constexpr int kB  = 4;
constexpr int kNQ = 900;
constexpr int kNC = 10;
constexpr int kH  = 200;
constexpr int kW  = 200;
constexpr int kHW = kH * kW;          // 40000
constexpr int kNA = 100;              // NUM_ANCHOR
constexpr int kNP = 900;              // NUM_POINTS
constexpr int kE  = 256;              // EMBED

// ---------------- workspace layout (bytes) ----------------
constexpr size_t OFF_SCORES = 0;                          // kB*kNQ floats   (14.4 KB)
constexpr size_t OFF_FILT   = 16384;                      // kB*kNA ints     (1.6 KB)
constexpr size_t OFF_COORDS = 20480;                      // kB*kNA*4 ints   (6.4 KB)
constexpr size_t OFF_MASK   = 32768;                      // kB*kHW ints     (640 KB)
constexpr size_t OFF_SEL    = OFF_MASK + (size_t)kB * kHW * sizeof(int); // kB*kHW bytes

typedef __attribute__((ext_vector_type(2))) float v2f;
typedef __attribute__((ext_vector_type(8))) float v8f;

// K1: scores = softmax(logits)[..., :-1].max(-1)  == exp(max9 - logsumexp10)
__global__ void k_scores(const float* __restrict__ logits, float* __restrict__ scores) {
    int idx = blockIdx.x * blockDim.x + threadIdx.x;
    if (idx >= kB * kNQ) return;
    const float* l = logits + (size_t)idx * kNC;
    float m = l[0];
#pragma unroll
    for (int c = 1; c < kNC; ++c) m = fmaxf(m, l[c]);
    float s = 0.f;
#pragma unroll
    for (int c = 0; c < kNC; ++c) s += __expf(l[c] - m);
    float m9 = l[0];
#pragma unroll
    for (int c = 1; c < kNC - 1; ++c) m9 = fmaxf(m9, l[c]);
    scores[idx] = __expf(m9 - m) / s;
}

// K2: exact top-k order via rank = #{j : s_j > s_q} + #{j < q : s_j == s_q}
// (matches jax.lax.top_k descending order with index tie-break; 900^2 cmp/batch)
__global__ void k_topk(const float* __restrict__ scores, int* __restrict__ filt) {
    __shared__ float s[kNQ];
    int b = blockIdx.x;
    for (int q = threadIdx.x; q < kNQ; q += blockDim.x) s[q] = scores[b * kNQ + q];
    __syncthreads();
    for (int q = threadIdx.x; q < kNQ; q += blockDim.x) {
        float v = s[q];
        int rank = 0;
        for (int j = 0; j < kNQ; ++j) {
            float u = s[j];
            rank += (u > v) || (u == v && j < q);
        }
        if (rank < kNA) filt[b * kNA + rank] = q;
    }
}

// K3: coords = int32( view5d(5x5) @ homog(5x100) ) via V_WMMA_F32_16X16X4_F32.
// One wave per batch. A = view5d padded to 16x4 (two K-chunks: j=0..3, then j=4).
// A layout (32-bit 16x4): lanes 0-15 hold (M=lane, K=0..1) in v[0..1],
// lanes 16-31 hold (M=lane-16, K=2..3). B/C/D row-striped across lanes.
__global__ void k_coords(const float* __restrict__ pred_boxes,
                         const float* __restrict__ view5d,
                         const int* __restrict__ filt,
                         int* __restrict__ coords) {
    int b    = blockIdx.x;
    int lane = threadIdx.x;        // 0..31, wave32
    int Mi   = lane & 15;
    int kh   = (lane >> 4) << 1;   // 0 or 2 (K pair held by this half-wave)
    const float* V = view5d + b * 25;

    v2f A1 = {0.f, 0.f}, A2 = {0.f, 0.f};
    if (Mi < 5) {                               // rows i = 0..4 of view5d
        A1.x = V[Mi * 5 + kh];                  // j = kh
        A1.y = V[Mi * 5 + kh + 1];              // j = kh+1
        if (kh == 0) A2.x = V[Mi * 5 + 4];      // j = 4 (second K-chunk, slot K=0)
    }

    for (int t = 0; t < (kNA + 15) / 16; ++t) {
        int n = t * 16 + Mi;                    // anchor column for this lane
        float hg0 = 0.f, hg1 = 0.f, hg2 = 0.f, hg3 = 0.f, hg4 = 0.f;
        if (n < kNA) {
            int q = filt[b * kNA + n];
            const float* pb = pred_boxes + ((size_t)b * kNQ + q) * 4;
            float cx = pb[0], cy = pb[1];
            float w = __expf(pb[2]), h = __expf(pb[3]);
            if (n == 2 || n == 3) {             // reference quirk: boxes.at[:, 2:4] exp'd again
                cx = __expf(cx); cy = __expf(cy); w = __expf(w); h = __expf(h);
            }
            hg0 = cx - 0.5f * w; hg1 = cy - 0.5f * h;
            hg2 = cx + 0.5f * w; hg3 = cy + 0.5f * h; hg4 = 1.f;
        }
        // B (4x16): VGPR0 rows K={0,2}, VGPR1 rows K={1,3} (lane halves as C/D pattern)
        v2f B1, B2;
        B1.x = (kh == 0) ? hg0 : hg2;
        B1.y = (kh == 0) ? hg1 : hg3;
        B2.x = (kh == 0) ? hg4 : 0.f;
        B2.y = 0.f;
        v8f C = {};
        C = __builtin_amdgcn_wmma_f32_16x16x4_f32(false, A1, false, B1, (short)0, C, false, false);
        C = __builtin_amdgcn_wmma_f32_16x16x4_f32(false, A2, false, B2, (short)0, C, false, false);
        // D: lanes 0-15 hold rows M=0..7 in C[0..7]; we need i=0..3 (drop homog row 4)
        if (lane < 16 && n < kNA) {
            int* o = coords + ((size_t)b * kNA + n) * 4;
            o[0] = (int)C[0]; o[1] = (int)C[1]; o[2] = (int)C[2]; o[3] = (int)C[3];
        }
    }
}

// K4: mask[b,p] = any over 100 boxes of (x in [x0,x1] && y in [y0,y1]); coords in LDS
__global__ void k_mask(const int* __restrict__ coords, int* __restrict__ mask) {
    __shared__ int4 cs[kNA];
    int b = blockIdx.y;
    for (int n = threadIdx.x; n < kNA; n += blockDim.x)
        cs[n] = ((const int4*)coords)[b * kNA + n];
    __syncthreads();
    int p = blockIdx.x * blockDim.x + threadIdx.x;
    if (p >= kHW) return;
    int x = p % kW, y = p / kW;
    int m = 0;
#pragma unroll 4
    for (int n = 0; n < kNA; ++n) {
        int4 c = cs[n];
        m |= (x >= c.x) & (x <= c.z) & (y >= c.y) & (y <= c.w);
    }
    mask[b * kHW + p] = m;
}

// K5: per-batch exclusive prefix scan of mask; sel = mask && rank < 900
// (stable argsort of where(mask, arange, HW+rnd) puts masked pixels first in
//  index order, so grid_idx == first 900 masked pixels in the M>=900 regime)
__global__ void k_select(const int* __restrict__ mask, unsigned char* __restrict__ sel) {
    __shared__ int sd[256];
    int b = blockIdx.x, tid = threadIdx.x;
    int running = 0;                     // identical across threads
    for (int base = 0; base < kHW; base += 256) {
        int p = base + tid;
        int m = (p < kHW) ? mask[b * kHW + p] : 0;
        sd[tid] = m;
        __syncthreads();
        for (int off = 1; off < 256; off <<= 1) {   // Hillis-Steele inclusive scan
            int t = (tid >= off) ? sd[tid - off] : 0;
            __syncthreads();
            sd[tid] += t;
            __syncthreads();
        }
        int incl  = sd[tid];
        int total = sd[255];
        int rank  = running + incl - m;             // exclusive rank
        if (p < kHW) sel[b * kHW + p] = (m && rank < kNP) ? 1 : 0;
        running += total;
        __syncthreads();                            // protect sd before next tile
    }
}

// K6: out[b,c,p] = sel ? query_weight[p,c] : mask_embedding[c]
// One block per (b,c): 164 MB of perfectly coalesced stores (store-BW bound,
// ~7us at 23.3 TB/s). sel (40 KB/batch as bytes) and query_weight (40 MB)
// stay L2-resident. Prefetch next tile's qw element when it will be selected.
__global__ void k_fill(const float* __restrict__ qw, const float* __restrict__ memb,
                       const unsigned char* __restrict__ sel, float* __restrict__ out) {
    int bc = blockIdx.x;                 // 0 .. kB*kE-1
    int b = bc / kE, c = bc % kE;
    float mval = memb[c];
    const unsigned char* sb = sel + (size_t)b * kHW;
    float* ob = out + (size_t)bc * kHW;
    for (int base = 0; base < kHW; base += 256) {
        int p = base + threadIdx.x;
        if (p < kHW) {
            int pn = p + 256;
            if (pn < kHW && sb[pn])
                __builtin_prefetch(qw + (size_t)pn * kE + c, 0, 0);  // global_prefetch_b8
            float v = sb[p] ? qw[(size_t)p * kE + c] : mval;
            ob[p] = v;
        }
    }
}

extern "C" void kernel_launch(void* const* d_in, const int* in_sizes, int n_in,
                              void* d_out, int out_size, void* d_ws, size_t ws_size,
                              hipStream_t stream) {
    const float* pred_boxes  = (const float*)d_in[0];
    const float* pred_logits = (const float*)d_in[1];
    const float* view5d      = (const float*)d_in[2];
    const float* qw          = (const float*)d_in[3];
    const float* memb        = (const float*)d_in[4];
    float* out = (float*)d_out;

    char* ws = (char*)d_ws;
    float*         scores = (float*)(ws + OFF_SCORES);
    int*           filt   = (int*)(ws + OFF_FILT);
    int*           coords = (int*)(ws + OFF_COORDS);
    int*           mask   = (int*)(ws + OFF_MASK);
    unsigned char* sel    = (unsigned char*)(ws + OFF_SEL);

    k_scores<<<(kB * kNQ + 255) / 256, 256, 0, stream>>>(pred_logits, scores);
    k_topk  <<<kB, 256, 0, stream>>>(scores, filt);
    k_coords<<<kB, 32, 0, stream>>>(pred_boxes, view5d, filt, coords);
    dim3 gm((kHW + 255) / 256, kB);
    k_mask  <<<gm, 256, 0, stream>>>(coords, mask);
    k_select<<<kB, 256, 0, stream>>>(mask, sel);
    k_fill  <<<kB * kE, 256, 0, stream>>>(qw, memb, sel, out);
}